// PointWiseDecoder2D_37443524887303
// MI455X (gfx1250) — compile-verified
//
#include <hip/hip_runtime.h>
#include <hip/hip_bf16.h>
#include <math.h>

typedef float v2f __attribute__((ext_vector_type(2)));
typedef float v8f __attribute__((ext_vector_type(8)));

#define ACT_NONE 0
#define ACT_SIN  1
#define ACT_GELU 2

__device__ __forceinline__ float gelu_exact(float x) {
    return 0.5f * x * (1.0f + erff(x * 0.70710678118654752f));
}

// ---------------------------------------------------------------------------
// Generic strided fp32 WMMA GEMM:
//   C[m,n] = epilogue( scale * sum_k A[m,k] * B[k,n] )
// A addr = A + z*sAz + m*sAm + k*sAk   (elements); AK1: sAk==1 (vector loads)
// B addr = B + z*sBz + k*sBk + n*sBn
// C addr = C + z*sCz + m*sCm + n*sCn
// Block: 128 threads = 4 waves. Each wave owns TWO 16x16 tiles (32 rows),
// so each LDS B fragment feeds 2 WMMAs. Block tile = 128 rows x 16 cols.
// K-chunks of 64 with double-buffered LDS B staging; A fragments for a whole
// chunk are preloaded into registers, then 32 v_wmma issue back-to-back.
// M must be a multiple of 128, K a multiple of 64; N is bounds-checked.
// ---------------------------------------------------------------------------
template <bool AK1>
__global__ void gemm_wmma_f32(
    const float* __restrict__ A, int sAm, int sAk, long long sAz,
    const float* __restrict__ B, int sBk, int sBn, long long sBz,
    float* __restrict__ C, int sCm, int sCn, long long sCz,
    const float* __restrict__ bias,
    const float* __restrict__ resid, int sRm, long long sRz,
    int N, int K, int act, float w0, float scale)
{
    __shared__ float Bs[2][64][17];

    const int lane = threadIdx.x & 31;
    const int wave = threadIdx.x >> 5;
    const int z    = blockIdx.z;
    const int row0 = blockIdx.x * 128 + wave * 32;  // two 16-row tiles per wave
    const int col0 = blockIdx.y * 16;

    const int mi = lane & 15;        // row within tile (A) / col within tile (B,C)
    const int kb = (lane >> 4) * 2;  // K sub-offset for A/B fragments

    v8f acc0 = {0.f, 0.f, 0.f, 0.f, 0.f, 0.f, 0.f, 0.f};
    v8f acc1 = {0.f, 0.f, 0.f, 0.f, 0.f, 0.f, 0.f, 0.f};

    const float* aRow0 = A + (long long)z * sAz + (long long)(row0 + mi) * sAm;
    const float* aRow1 = aRow0 + (long long)16 * sAm;

    // B staging: thread -> (skk, snn); j loop advances skk by 8
    const int  skk  = threadIdx.x >> 4;   // 0..7
    const int  snn  = threadIdx.x & 15;
    const int  sn   = col0 + snn;
    const bool snok = (sn < N);
    const float* bBase = B + (long long)z * sBz + (long long)sn * sBn
                           + (long long)skk * sBk;
    const long long bStepJ = 8LL * sBk;
    const long long bStepC = 64LL * sBk;

    float breg[8];
    #pragma unroll
    for (int j = 0; j < 8; ++j)
        breg[j] = snok ? bBase[j * bStepJ] : 0.f;
    #pragma unroll
    for (int j = 0; j < 8; ++j)
        Bs[0][skk + j * 8][snn] = breg[j];
    __syncthreads();

    const int nChunks = K >> 6;
    for (int ch = 0; ch < nChunks; ++ch) {
        const int  k0   = ch << 6;
        const int  buf  = ch & 1;
        const bool more = (ch + 1) < nChunks;

        // issue next chunk's B global loads early (latency overlapped below)
        if (more) {
            const float* bp = bBase + (long long)(ch + 1) * bStepC;
            #pragma unroll
            for (int j = 0; j < 8; ++j)
                breg[j] = snok ? bp[j * bStepJ] : 0.f;
        }

        // preload the whole chunk's A fragments into registers
        v2f a0[16], a1[16];
        if (AK1) {
            const float* p0 = aRow0 + k0 + kb;
            const float* p1 = aRow1 + k0 + kb;
            #pragma unroll
            for (int ks = 0; ks < 16; ++ks) {
                a0[ks] = *(const v2f*)(p0 + ks * 4);
                a1[ks] = *(const v2f*)(p1 + ks * 4);
            }
        } else {
            #pragma unroll
            for (int ks = 0; ks < 16; ++ks) {
                const long long kk = (long long)(k0 + ks * 4 + kb) * sAk;
                a0[ks].x = aRow0[kk];       a0[ks].y = aRow0[kk + sAk];
                a1[ks].x = aRow1[kk];       a1[ks].y = aRow1[kk + sAk];
            }
        }

        // 32 back-to-back WMMAs (two independent accumulator chains)
        #pragma unroll
        for (int ks = 0; ks < 16; ++ks) {
            v2f b;
            b.x = Bs[buf][ks * 4 + kb + 0][mi];
            b.y = Bs[buf][ks * 4 + kb + 1][mi];
            acc0 = __builtin_amdgcn_wmma_f32_16x16x4_f32(
                false, a0[ks], false, b, (short)0, acc0, false, false);
            acc1 = __builtin_amdgcn_wmma_f32_16x16x4_f32(
                false, a1[ks], false, b, (short)0, acc1, false, false);
        }

        // park next chunk's B into the other LDS buffer
        if (more) {
            #pragma unroll
            for (int j = 0; j < 8; ++j)
                Bs[1 - buf][skk + j * 8][snn] = breg[j];
        }
        __syncthreads();
    }

    // epilogue: scale, +bias, activation, +residual, strided store
    const int  n   = col0 + mi;
    const bool nok = (n < N);
    const float bv = (bias != nullptr && nok) ? bias[n] : 0.f;
    const int  mhi = (lane >> 4) * 8;
    #pragma unroll
    for (int t = 0; t < 2; ++t) {
        const v8f acc = t ? acc1 : acc0;
        const int rbase = row0 + t * 16;
        #pragma unroll
        for (int i = 0; i < 8; ++i) {
            const int m = rbase + i + mhi;
            float v = acc[i] * scale + bv;
            if (act == ACT_SIN)       v = sinf(w0 * v);
            else if (act == ACT_GELU) v = gelu_exact(v);
            if (nok) {
                if (resid != nullptr)
                    v += resid[(long long)z * sRz + (long long)m * sRm + n];
                C[(long long)z * sCz + (long long)m * sCm + (long long)n * sCn] = v;
            }
        }
    }
}

// ---------------------------------------------------------------------------
// Per-batch min/max of each of the 2 coordinate dims over Nq=8192 queries.
// stats[b*4 + {0,1,2,3}] = mn0, mx0, mn1, mx1
// ---------------------------------------------------------------------------
__global__ void minmax_pos(const float* __restrict__ pos, float* __restrict__ stats)
{
    const int b = blockIdx.x;
    const int t = threadIdx.x;
    const float* p = pos + (long long)b * 8192 * 2;
    float mn0 = 1e30f, mx0 = -1e30f, mn1 = 1e30f, mx1 = -1e30f;
    for (int i = t; i < 8192; i += 256) {
        const float a = p[i * 2 + 0];
        const float c = p[i * 2 + 1];
        mn0 = fminf(mn0, a); mx0 = fmaxf(mx0, a);
        mn1 = fminf(mn1, c); mx1 = fmaxf(mx1, c);
    }
    __shared__ float s0[256], s1[256], s2[256], s3[256];
    s0[t] = mn0; s1[t] = mx0; s2[t] = mn1; s3[t] = mx1;
    __syncthreads();
    for (int o = 128; o > 0; o >>= 1) {
        if (t < o) {
            s0[t] = fminf(s0[t], s0[t + o]);
            s1[t] = fmaxf(s1[t], s1[t + o]);
            s2[t] = fminf(s2[t], s2[t + o]);
            s3[t] = fmaxf(s3[t], s3[t + o]);
        }
        __syncthreads();
    }
    if (t == 0) {
        stats[b * 4 + 0] = s0[0];
        stats[b * 4 + 1] = s1[0];
        stats[b * 4 + 2] = s2[0];
        stats[b * 4 + 3] = s3[0];
    }
}

// ---------------------------------------------------------------------------
// SIREN layer 0: coord normalize, 2->256 matmul, sin(30*.)  (K=2, VALU)
// out: (32768, 256)
// ---------------------------------------------------------------------------
__global__ void siren_layer0(const float* __restrict__ pos,
                             const float* __restrict__ stats,
                             const float* __restrict__ W0,   // (2,256) row-major
                             const float* __restrict__ b0,   // (256)
                             float* __restrict__ out)
{
    const int gid = blockIdx.x * blockDim.x + threadIdx.x;
    const int row = gid >> 8;
    const int c   = gid & 255;
    const int b   = row >> 13;
    const float mn0 = stats[b * 4 + 0], mx0 = stats[b * 4 + 1];
    const float mn1 = stats[b * 4 + 2], mx1 = stats[b * 4 + 3];
    const float p0 = pos[(long long)row * 2 + 0];
    const float p1 = pos[(long long)row * 2 + 1];
    const float x0 = (2.f * p0 - mn0 - mx0) / (mx0 - mn0);
    const float x1 = (2.f * p1 - mn1 - mx1) / (mx1 - mn1);
    const float h = x0 * W0[c] + x1 * W0[256 + c] + b0[c];
    out[(long long)row * 256 + c] = sinf(30.f * h);
}

// ---------------------------------------------------------------------------
// 2D rotary embedding, in place, pairwise rotation.
// Per row: 4 heads * 2 halves (x/y pos) * 64 rotation pairs = 512 work items.
// rowStride=1024 for q buffer, 2048 for kv (rotates k cols 0..1023 only).
// ---------------------------------------------------------------------------
__global__ void rotary2d(float* __restrict__ t,
                         const float* __restrict__ pos,
                         int rowStride, int nRows)
{
    const int gid = blockIdx.x * blockDim.x + threadIdx.x;
    if (gid >= nRows * 512) return;
    const int row  = gid >> 9;
    const int rem  = gid & 511;
    const int h    = rem >> 7;
    const int half = (rem >> 6) & 1;
    const int i    = rem & 63;
    const float p    = pos[(long long)row * 2 + half];
    // inv_freq[i] = 10000^(-2i/128);  SCALE/MIN_FREQ = 16*64 = 1024
    const float invf = expf(-(2.f * (float)i / 128.f) * 9.210340371976184f);
    const float ang  = p * 1024.f * invf;
    float sn, cs;
    sincosf(ang, &sn, &cs);
    const long long base = (long long)row * rowStride + h * 256 + half * 128 + i;
    const float a = t[base];
    const float b = t[base + 64];
    t[base]      = a * cs - b * sn;   // t*cos + rot_half(t)*sin, j<64 half
    t[base + 64] = b * cs + a * sn;   // j>=64 half
}

// ---------------------------------------------------------------------------
// Instance norm over the token axis (4096 rows) for each (b, col) column of
// the kv buffer (B, 4096, 2048). Block = 32 cols x 8 row-groups (coalesced).
// Grid = B * 64.
// ---------------------------------------------------------------------------
__global__ void inorm_cols(float* __restrict__ kv)
{
    const int b  = blockIdx.x >> 6;
    const int cg = blockIdx.x & 63;
    const int c  = threadIdx.x & 31;
    const int rg = threadIdx.x >> 5;   // 0..7
    float* base = kv + (long long)b * 4096 * 2048 + cg * 32 + c;
    float s = 0.f, sq = 0.f;
    for (int r = rg; r < 4096; r += 8) {
        const float v = base[(long long)r * 2048];
        s += v; sq += v * v;
    }
    __shared__ float sh_s[8][32], sh_q[8][32];
    __shared__ float sh_mean[32], sh_rstd[32];
    sh_s[rg][c] = s; sh_q[rg][c] = sq;
    __syncthreads();
    if (rg == 0) {
        float ts = 0.f, tq = 0.f;
        #pragma unroll
        for (int i = 0; i < 8; ++i) { ts += sh_s[i][c]; tq += sh_q[i][c]; }
        const float mean = ts * (1.f / 4096.f);
        const float var  = tq * (1.f / 4096.f) - mean * mean;
        sh_mean[c] = mean;
        sh_rstd[c] = rsqrtf(var + 1e-5f);
    }
    __syncthreads();
    const float mean = sh_mean[c], rstd = sh_rstd[c];
    for (int r = rg; r < 4096; r += 8) {
        const long long idx = (long long)r * 2048;
        base[idx] = (base[idx] - mean) * rstd;
    }
}

// ---------------------------------------------------------------------------

static void launch_gemm(hipStream_t st,
    const float* A, int sAm, int sAk, long long sAz,
    const float* B, int sBk, int sBn, long long sBz,
    float* C, int sCm, int sCn, long long sCz,
    const float* bias, const float* resid, int sRm, long long sRz,
    int M, int N, int K, int Z, int act, float w0, float scale)
{
    dim3 grid(M / 128, (N + 15) / 16, Z);
    if (sAk == 1)
        gemm_wmma_f32<true><<<grid, 128, 0, st>>>(A, sAm, sAk, sAz, B, sBk, sBn, sBz,
                                                  C, sCm, sCn, sCz, bias, resid, sRm, sRz,
                                                  N, K, act, w0, scale);
    else
        gemm_wmma_f32<false><<<grid, 128, 0, st>>>(A, sAm, sAk, sAz, B, sBk, sBn, sBz,
                                                   C, sCm, sCn, sCz, bias, resid, sRm, sRz,
                                                   N, K, act, w0, scale);
}

extern "C" void kernel_launch(void* const* d_in, const int* in_sizes, int n_in,
                              void* d_out, int out_size, void* d_ws, size_t ws_size,
                              hipStream_t stream) {
    (void)in_sizes; (void)n_in; (void)out_size; (void)ws_size;

    // inputs in setup_inputs() dict order (tuples flattened as leaves)
    const float* z    = (const float*)d_in[0];   // (4, 4096, 256)
    const float* ppos = (const float*)d_in[1];   // (4, 8192, 2)
    const float* ipos = (const float*)d_in[2];   // (4, 4096, 2)
    const float* sw[5]; const float* sb[5];
    for (int i = 0; i < 5; ++i) { sw[i] = (const float*)d_in[3 + i]; sb[i] = (const float*)d_in[8 + i]; }
    const float* w_kv = (const float*)d_in[13];  // (256, 2048)
    const float* w_ao = (const float*)d_in[14];  // (1024, 256)
    const float* b_ao = (const float*)d_in[15];  // (256)
    const float* fw1  = (const float*)d_in[16];
    const float* fb1  = (const float*)d_in[17];
    const float* fw2  = (const float*)d_in[18];
    const float* fb2  = (const float*)d_in[19];
    const float* dw1  = (const float*)d_in[20];  // (256, 128)
    const float* dw2  = (const float*)d_in[21];  // (128, 128)
    const float* dw3  = (const float*)d_in[22];  // (128, 10)
    const float* db3  = (const float*)d_in[23];  // (10)

    float* out  = (float*)d_out;        // u: (4, 10, 8192) = 327680 floats
    float* xout = out + 327680;         // x: (4, 8192, 256) = 8388608 floats

    // workspace layout (floats)
    float* ws    = (float*)d_ws;
    float* q     = ws;                    // (32768, 1024)
    float* kv    = q    + 33554432LL;     // (16384, 2048); reused as attn out
    float* s0    = kv   + 33554432LL;     // (32768, 256)
    float* s1    = s0   + 8388608LL;      // (32768, 256)
    float* dots  = s1   + 8388608LL;      // (16, 256, 256)
    float* stats = dots + 1048576LL;      // (4, 4)

    // ---- SIREN ----
    minmax_pos<<<4, 256, 0, stream>>>(ppos, stats);
    siren_layer0<<<32768, 256, 0, stream>>>(ppos, stats, sw[0], sb[0], s0);
    // hidden layers: 256->256 with sin(1.0 * .)
    launch_gemm(stream, s0,256,1,0, sw[1],256,1,0, s1,256,1,0, sb[1], nullptr,0,0,
                32768,256,256,1, ACT_SIN, 1.f, 1.f);
    launch_gemm(stream, s1,256,1,0, sw[2],256,1,0, s0,256,1,0, sb[2], nullptr,0,0,
                32768,256,256,1, ACT_SIN, 1.f, 1.f);
    launch_gemm(stream, s0,256,1,0, sw[3],256,1,0, s1,256,1,0, sb[3], nullptr,0,0,
                32768,256,256,1, ACT_SIN, 1.f, 1.f);
    // output layer: 256->1024, no activation -> q
    launch_gemm(stream, s1,256,1,0, sw[4],1024,1,0, q,1024,1,0, sb[4], nullptr,0,0,
                32768,1024,256,1, ACT_NONE, 0.f, 1.f);

    // ---- kv projection: z @ w_kv  (16384 x 2048) ----
    launch_gemm(stream, z,256,1,0, w_kv,2048,1,0, kv,2048,1,0, nullptr, nullptr,0,0,
                16384,2048,256,1, ACT_NONE, 0.f, 1.f);

    // ---- rotary on q (propagate_pos) and on k half of kv (input_pos) ----
    rotary2d<<<(32768 * 512) / 256, 256, 0, stream>>>(q,  ppos, 1024, 32768);
    rotary2d<<<(16384 * 512) / 256, 256, 0, stream>>>(kv, ipos, 2048, 16384);

    // ---- instance norm of k and v over the token axis ----
    inorm_cols<<<4 * 64, 256, 0, stream>>>(kv);

    // ---- dots[b,h] = k^T v : (256 x 4096) @ (4096 x 256) ----
    for (int b = 0; b < 4; ++b)
        for (int h = 0; h < 4; ++h) {
            const float* Ak = kv + (long long)b * 8388608 + h * 256;          // k: A[m,kk]
            const float* Bv = kv + (long long)b * 8388608 + 1024 + h * 256;   // v
            float* Cd = dots + (long long)(b * 4 + h) * 65536;
            launch_gemm(stream, Ak,1,2048,0, Bv,2048,1,0, Cd,256,1,0,
                        nullptr, nullptr,0,0, 256,256,4096,1, ACT_NONE, 0.f, 1.f);
        }

    // ---- out[b,h] = q @ dots / Nkv : (8192 x 256) @ (256 x 256) -> attn (reuse kv) ----
    for (int b = 0; b < 4; ++b)
        for (int h = 0; h < 4; ++h) {
            const float* Aq = q + (long long)b * 8388608 + h * 256;
            const float* Bd = dots + (long long)(b * 4 + h) * 65536;
            float* Ca = kv + (long long)b * 8388608 + h * 256;
            launch_gemm(stream, Aq,1024,1,0, Bd,256,1,0, Ca,1024,1,0,
                        nullptr, nullptr,0,0, 8192,256,256,1, ACT_NONE, 0.f, 1.f / 4096.f);
        }

    // ---- attention output projection: (32768 x 1024) @ (1024 x 256) + b -> s0 ----
    launch_gemm(stream, kv,1024,1,0, w_ao,256,1,0, s0,256,1,0, b_ao, nullptr,0,0,
                32768,256,1024,1, ACT_NONE, 0.f, 1.f);

    // ---- FFN with residual, writing x directly into d_out ----
    launch_gemm(stream, s0,256,1,0, fw1,256,1,0, s1,256,1,0, fb1, nullptr,0,0,
                32768,256,256,1, ACT_GELU, 0.f, 1.f);
    launch_gemm(stream, s1,256,1,0, fw2,256,1,0, xout,256,1,0, fb2, s0,256,0,
                32768,256,256,1, ACT_NONE, 0.f, 1.f);

    // ---- decoder ----
    launch_gemm(stream, xout,256,1,0, dw1,128,1,0, s0,128,1,0, nullptr, nullptr,0,0,
                32768,128,256,1, ACT_GELU, 0.f, 1.f);
    launch_gemm(stream, s0,128,1,0, dw2,128,1,0, s1,128,1,0, nullptr, nullptr,0,0,
                32768,128,128,1, ACT_GELU, 0.f, 1.f);
    // final 128->10, written transposed: u[b, n, m] = d_out[b*81920 + n*8192 + m]
    launch_gemm(stream, s1,128,1, (long long)8192 * 128,
                dw3,10,1, 0,
                out,1,8192, (long long)81920,
                db3, nullptr,0,0, 8192,10,128,4, ACT_NONE, 0.f, 1.f);
}